// ROEWAGradient_79671643341080
// MI455X (gfx1250) — compile-verified
//
#include <hip/hip_runtime.h>
#include <hip/hip_bf16.h>

// ROEWA gradient for (8,3,1024,1024) f32.
// Memory-bound (~2 GB total traffic over 8 streaming passes -> ~86us @ 23.3TB/s).
// H-direction scans: 1 thread/column VALU streaming (perfectly coalesced).
// W-direction scans: scan-as-matmul on V_WMMA_F32_16X16X4_F32 (f32 precision),
// tiles staged via TDM tensor_load_to_lds with LDS padding (pitch 66) and
// s_wait_tensorcnt; carry handed between 16-wide chunks via wave shuffles.

#define ALPHA_F 0.9f
#define EPS_F   1e-12f
#define LOG2A   (-0.15200309344504997f)   // log2(0.9)

static constexpr int       HH     = 1024;
static constexpr int       WW     = 1024;
static constexpr int       PLANES = 8 * 3;
static constexpr long long NELEM  = (long long)PLANES * HH * WW;

// row-scan geometry
static constexpr int TILE_W = 64;   // TDM tile width (f32 elems)
static constexpr int PITCH  = 66;   // TDM pad: +2 dwords per 64 -> bank-conflict-free
static constexpr int RPW    = 16;   // rows per wave = WMMA tile height
static constexpr int NWAVES = 4;    // waves per block (keeps static LDS < 64KB)
static constexpr int ROWS_PER_BLOCK = RPW * NWAVES;        // 64
static constexpr int ROW_BLOCKS = (PLANES * HH) / ROWS_PER_BLOCK; // 384
static constexpr int COL_BLOCKS = (PLANES * WW) / 256;            // 96

typedef __attribute__((ext_vector_type(2))) float        v2f;
typedef __attribute__((ext_vector_type(8))) float        v8f;
typedef __attribute__((ext_vector_type(4))) unsigned int v4u;
typedef __attribute__((ext_vector_type(8))) int          v8i;
typedef __attribute__((ext_vector_type(4))) int          v4i;

__device__ __forceinline__ float apow(int e) {          // ALPHA^e, e>=0
  return exp2f(LOG2A * (float)e);
}

// ---------------------------------------------------------------------------
// TDM: DMA one 16 x 64 f32 tile (row stride WW) from global into LDS with
// +2-dword padding every row (pad_interval=64dw, pad_amount=2dw -> pitch 66).
// ---------------------------------------------------------------------------
__device__ __forceinline__ void tdm_tile_to_lds(const float* gsrc, float* lds_dst) {
#if __has_builtin(__builtin_amdgcn_tensor_load_to_lds)
  unsigned long long ga = (unsigned long long)(uintptr_t)gsrc;
  unsigned galo = (unsigned)__builtin_amdgcn_readfirstlane((int)(unsigned)(ga & 0xffffffffull));
  unsigned gahi = (unsigned)__builtin_amdgcn_readfirstlane((int)(unsigned)(ga >> 32));
  unsigned loff = (unsigned)__builtin_amdgcn_readfirstlane((int)(unsigned)(uintptr_t)lds_dst);
  v4u g0;
  g0.x = 1u;                                  // count=1 valid user descriptor
  g0.y = loff;                                // lds_addr
  g0.z = galo;                                // global_addr[31:0] = tile start
  g0.w = (gahi & 0x01ffffffu) | (2u << 30);   // global_addr[56:32] | type=2
  v8i g1;
  g1[0] = 0x03520000;                         // data_size=4B | pad_en | ivl=64dw | amt=2dw
  g1[1] = (WW & 0xffff) << 16;                // tensor_dim0 lo16 (atomic addr = 0)
  g1[2] = (WW >> 16) | ((HH & 0xffff) << 16); // tensor_dim0 hi | tensor_dim1 lo
  g1[3] = (HH >> 16) | (TILE_W << 16);        // tensor_dim1 hi | tile_dim0
  g1[4] = RPW;                                // tile_dim1 | tile_dim2=0
  g1[5] = WW;                                 // tensor_dim0_stride lo32
  g1[6] = 0;
  g1[7] = 0;
  v4i gz4 = {0, 0, 0, 0};                     // 2D tensor: groups 2/3 unused
  v8i gz8 = {0, 0, 0, 0, 0, 0, 0, 0};         // extra group (6-arg toolchain form)
  __builtin_amdgcn_tensor_load_to_lds(g0, g1, gz4, gz4, gz8, 0);
#else
  const int lane = threadIdx.x & 31;
  for (int i = lane; i < RPW * TILE_W; i += 32) {
    int r = i >> 6, c = i & 63;
    lds_dst[r * PITCH + c] = gsrc[(long long)r * WW + c];
  }
#endif
}

__device__ __forceinline__ void tdm_wait() {
#if __has_builtin(__builtin_amdgcn_s_wait_tensorcnt)
  __builtin_amdgcn_s_wait_tensorcnt((short)0);
#endif
}

// ---------------------------------------------------------------------------
// Row-direction scan kernel (W axis).  MODE: 0 = backward EMA,
// 1 = forward EMA + ISEF combine with src1 (=backward EMA),
// 2 = forward EMA + log-ratio against src1 (=backward EMA).
// One wave owns 16 rows; per 16-wide chunk: y(16x16) = T(16x16) x X(16x16)
// + carry outer-product, via 4 chained V_WMMA_F32_16X16X4_F32.
// ---------------------------------------------------------------------------
template <int MODE>
__global__ __launch_bounds__(NWAVES * 32)
void rowScanKernel(const float* __restrict__ src0,
                   const float* __restrict__ src1,
                   float* __restrict__ dst) {
  __shared__ float ldsA[NWAVES][RPW][PITCH];  // src0 tiles (WMMA input)
  __shared__ float ldsB[NWAVES][RPW][PITCH];  // src1 tiles (combine input)
  __shared__ float ldsO[NWAVES][RPW][PITCH];  // output staging

  const int tid  = threadIdx.x;
  const int wave = tid >> 5;
  const int lane = tid & 31;
  const int hi   = lane >> 4;   // half-wave: selects K pair (A/B) and M+8 (C/D)
  const int ln15 = lane & 15;   // A: row m; B/C/D: column n

  const long long row0 = (long long)blockIdx.x * ROWS_PER_BLOCK + (long long)wave * RPW;
  const float* s0 = src0 + row0 * WW;
  const float* s1 = (MODE >= 1) ? (src1 + row0 * WW) : nullptr;
  float* d0 = dst + row0 * WW;

  constexpr bool FWD = (MODE != 0);
  const float onema  = 1.0f - ALPHA_F;
  const float inv1pa = 1.0f / (1.0f + ALPHA_F);

  // --- A fragments: T[m][k] chunks of K=4.  Lane layout (f32 16x4 A):
  // lanes 0-15: m=0..15, {VGPR0:K=k0, VGPR1:K=k0+1}; lanes 16-31: K=k0+2,k0+3.
  v2f aMid[4], aEdge[4];
#pragma unroll
  for (int c2 = 0; c2 < 4; ++c2) {
    const int k0 = c2 * 4 + hi * 2;
    float wx, wy;
    if (FWD) {  // lower-triangular: w = (1-a) a^(m-k)
      wx = (ln15 >= k0)     ? onema * apow(ln15 - k0)     : 0.0f;
      wy = (ln15 >= k0 + 1) ? onema * apow(ln15 - k0 - 1) : 0.0f;
    } else {    // upper-triangular: w = (1-a) a^(k-m)
      wx = (k0 >= ln15)     ? onema * apow(k0 - ln15)     : 0.0f;
      wy = (k0 + 1 >= ln15) ? onema * apow(k0 + 1 - ln15) : 0.0f;
    }
    aMid[c2].x = wx;  aMid[c2].y = wy;
    aEdge[c2]  = aMid[c2];
    if (FWD) { if (k0 == 0)      aEdge[c2].x = apow(ln15); }      // y0 = x0 boundary
    else     { if (k0 + 1 == 15) aEdge[c2].y = apow(15 - ln15); } // y[N-1]=x[N-1]
  }
  // carry multiplier per C/D VGPR v (m = v + 8*hi)
  float cmul[8];
#pragma unroll
  for (int v = 0; v < 8; ++v) {
    const int m = v + 8 * hi;
    cmul[v] = FWD ? apow(m + 1) : apow(16 - m);
  }

  float carry = 0.0f;
  const int NT = WW / TILE_W;  // 16 tiles per row
  for (int ti = 0; ti < NT; ++ti) {
    const int t  = FWD ? ti : (NT - 1 - ti);
    const int w0 = t * TILE_W;

    tdm_tile_to_lds(s0 + w0, &ldsA[wave][0][0]);
    if (MODE >= 1) tdm_tile_to_lds(s1 + w0, &ldsB[wave][0][0]);
    tdm_wait();
    __syncthreads();  // order TDM LDS writes vs ds reads (uniform across block)

#pragma unroll
    for (int ci = 0; ci < 4; ++ci) {
      const int cc   = FWD ? ci : 3 - ci;
      const int ccol = cc * 16;
      const bool edge = FWD ? (t == 0 && cc == 0) : (t == NT - 1 && cc == 3);

      // B fragments: B[k][n] = X tile[row n][col ccol+k]
      v2f b[4];
#pragma unroll
      for (int c2 = 0; c2 < 4; ++c2) {
        const int k = ccol + c2 * 4 + hi * 2;
        b[c2].x = ldsA[wave][ln15][k];
        b[c2].y = ldsA[wave][ln15][k + 1];
      }

      // accumulator C = carry outer product: C[m][n] = a^(...) * carry[n]
      const float cv = __shfl(carry, (FWD ? 16 : 0) + ln15, 32);
      v8f acc;
#pragma unroll
      for (int v = 0; v < 8; ++v) acc[v] = cmul[v] * cv;

#pragma unroll
      for (int c2 = 0; c2 < 4; ++c2) {
        const v2f af = edge ? aEdge[c2] : aMid[c2];
        acc = __builtin_amdgcn_wmma_f32_16x16x4_f32(
            /*neg_a=*/false, af, /*neg_b=*/false, b[c2],
            /*c_mod=*/(short)0, acc, /*reuse_a=*/false, /*reuse_b=*/false);
      }

      // post-process + stage to LDS out tile (D: lane ln15=n, m = v + 8*hi)
#pragma unroll
      for (int v = 0; v < 8; ++v) {
        const int m = v + 8 * hi;
        const float y = acc[v];
        float o;
        if (MODE == 0) {
          o = y;
        } else if (MODE == 1) {
          const float xv = ldsA[wave][ln15][ccol + m];
          const float bv = ldsB[wave][ln15][ccol + m];
          o = (y + bv - onema * xv) * inv1pa;             // ISEF combine
        } else {
          const float bv = ldsB[wave][ln15][ccol + m];
          o = logf(fmaxf(y, 0.0f) + EPS_F) - logf(fmaxf(bv, 0.0f) + EPS_F);
        }
        ldsO[wave][ln15][ccol + m] = o;
      }
      carry = FWD ? acc[7] : acc[0];  // y[15][n] (lanes 16-31) / y[0][n] (lanes 0-15)
    }

    // coalesced store: each lane writes 2 consecutive floats per row
#pragma unroll 4
    for (int rr = 0; rr < RPW; ++rr) {
      float2 val;
      val.x = ldsO[wave][rr][2 * lane];
      val.y = ldsO[wave][rr][2 * lane + 1];
      *(float2*)(d0 + (long long)rr * WW + w0 + 2 * lane) = val;
    }
    __syncthreads();
  }
}

// ---------------------------------------------------------------------------
// Column-direction kernels (H axis): one thread per column, fully coalesced.
// ---------------------------------------------------------------------------
__global__ __launch_bounds__(256)
void colBwdEma(const float* __restrict__ src, float* __restrict__ dst) {
  const long long cg = (long long)blockIdx.x * 256 + threadIdx.x;
  const long long base = (cg >> 10) * (long long)HH * WW + (cg & 1023);
  const float* s = src + base;
  float* d = dst + base;
  float y = s[(long long)(HH - 1) * WW];
  d[(long long)(HH - 1) * WW] = y;
  for (int r = HH - 2; r >= 0; --r) {
    const float x = s[(long long)r * WW];
    y = ALPHA_F * y + (1.0f - ALPHA_F) * x;
    d[(long long)r * WW] = y;
  }
}

template <int MODE>  // 1 = ISEF combine, 2 = log-ratio
__global__ __launch_bounds__(256)
void colFwdKernel(const float* __restrict__ src0, const float* __restrict__ src1,
                  float* __restrict__ dst) {
  const long long cg = (long long)blockIdx.x * 256 + threadIdx.x;
  const long long base = (cg >> 10) * (long long)HH * WW + (cg & 1023);
  const float* s0 = src0 + base;
  const float* s1 = src1 + base;
  float* d = dst + base;
  const float onema = 1.0f - ALPHA_F;
  const float inv1pa = 1.0f / (1.0f + ALPHA_F);
  float y = 0.0f;
  for (int r = 0; r < HH; ++r) {
    const long long o = (long long)r * WW;
    const float x = s0[o];
    y = (r == 0) ? x : (ALPHA_F * y + onema * x);
    const float b = s1[o];
    if (MODE == 1) d[o] = (y + b - onema * x) * inv1pa;
    else           d[o] = logf(fmaxf(y, 0.0f) + EPS_F) - logf(fmaxf(b, 0.0f) + EPS_F);
  }
}

// ---------------------------------------------------------------------------
// Launch.  Needs d_ws >= NELEM*4 bytes (100.7 MB): one intermediate plane-set.
// The not-yet-final gy half of d_out doubles as the second scratch buffer.
// ---------------------------------------------------------------------------
extern "C" void kernel_launch(void* const* d_in, const int* in_sizes, int n_in,
                              void* d_out, int out_size, void* d_ws, size_t ws_size,
                              hipStream_t stream) {
  (void)in_sizes; (void)n_in; (void)out_size; (void)ws_size;
  const float* img = (const float*)d_in[0];
  float* gx = (float*)d_out;
  float* gy = (float*)d_out + NELEM;
  float* ws = (float*)d_ws;

  // gx path: img_y = isef_H(img); sL/sR = ema_W(img_y); gx = logratio
  colBwdEma            <<<COL_BLOCKS, 256,        0, stream>>>(img, ws);       // ws = bwdH(img)
  colFwdKernel<1>      <<<COL_BLOCKS, 256,        0, stream>>>(img, ws, ws);   // ws = img_y
  rowScanKernel<0>     <<<ROW_BLOCKS, NWAVES * 32, 0, stream>>>(ws, nullptr, gy); // gy = sR
  rowScanKernel<2>     <<<ROW_BLOCKS, NWAVES * 32, 0, stream>>>(ws, gy, gx);      // gx done
  // gy path: img_x = isef_W(img); sU/sD = ema_H(img_x); gy = logratio
  rowScanKernel<0>     <<<ROW_BLOCKS, NWAVES * 32, 0, stream>>>(img, nullptr, ws); // ws = bwdW(img)
  rowScanKernel<1>     <<<ROW_BLOCKS, NWAVES * 32, 0, stream>>>(img, ws, ws);      // ws = img_x
  colBwdEma            <<<COL_BLOCKS, 256,        0, stream>>>(ws, gy);        // gy = sD
  colFwdKernel<2>      <<<COL_BLOCKS, 256,        0, stream>>>(ws, gy, gy);    // gy done
}